// GCNModel_24507083391315
// MI455X (gfx1250) — compile-verified
//
#include <hip/hip_runtime.h>
#include <hip/hip_bf16.h>
#include <math.h>

// ---------------------------------------------------------------------------
// GCN (3-layer) for MI455X / gfx1250, wave32.
//  - GEMMs use V_WMMA_F32_16X16X4_F32 (full fp32, matches reference numerics;
//    GEMM is only ~1.5 GFLOP so fp32 matrix path is the right choice).
//  - Edge aggregation = gather + global_atomic_add_f32 (L2-resident working set).
//  - Self-loop contribution fused into bias/ReLU pass (no atomics needed).
// ---------------------------------------------------------------------------

typedef __attribute__((ext_vector_type(2))) float v2f;
typedef __attribute__((ext_vector_type(8))) float v8f;

// ---------------------------- small utility kernels ------------------------

__global__ __launch_bounds__(256) void k_fill(float* __restrict__ p, float v, int n) {
  int i = blockIdx.x * blockDim.x + threadIdx.x;
  if (i < n) p[i] = v;
}

__global__ __launch_bounds__(256) void k_deg_accum(const long long* __restrict__ dst,
                                                   float* __restrict__ deg, int nE) {
  int e = blockIdx.x * blockDim.x + threadIdx.x;
  if (e < nE) atomicAdd(&deg[(int)dst[e]], 1.0f);
}

__global__ __launch_bounds__(256) void k_rsqrt_inplace(float* __restrict__ p, int n) {
  int i = blockIdx.x * blockDim.x + threadIdx.x;
  if (i < n) {
    float d = p[i];
    p[i] = (d > 0.0f) ? rsqrtf(d) : 0.0f;
  }
}

// ---------------------------- WMMA fp32 GEMM -------------------------------
// H[nRows, CO] = A[nRows, K] @ W[K, CO]
// One wave computes one 16x16 output tile; W is staged (zero-padded to
// CO_PAD cols) in LDS; A fragments are 8-byte global loads.
//
// V_WMMA_F32_16X16X4_F32 operand layouts (ISA 7.12.2, 32-bit A 16x4):
//   A:  lane L (m = L&15): vgpr j holds A[m][2*(L>>4) + j]
//   B:  lane L (n = L&15): vgpr j holds B[2*(L>>4) + j][n]
//   C/D:lane L (n = L&15): vgpr j holds D[8*(L>>4) + j][n]

template <int K, int CO, int CO_PAD>
__global__ __launch_bounds__(256) void k_gemm_wmma(const float* __restrict__ A,
                                                   const float* __restrict__ W,
                                                   float* __restrict__ H, int nRows) {
  __shared__ float ldsW[K * CO_PAD];
  for (int i = threadIdx.x; i < K * CO_PAD; i += 256) {
    int k = i / CO_PAD;
    int c = i - k * CO_PAD;
    ldsW[i] = (c < CO) ? W[k * CO + c] : 0.0f;
  }
  __syncthreads();

  const int lane = threadIdx.x & 31;
  const int wave = threadIdx.x >> 5;
  const int half = lane >> 4;   // selects K-pair {0,1} vs {2,3}
  const int mr   = lane & 15;

  const int rowTiles = nRows >> 4;        // nRows % 16 == 0 (N = 50000)
  const int colTiles = CO_PAD / 16;
  const int tile = blockIdx.x * 8 + wave; // 8 waves / block
  if (tile >= rowTiles * colTiles) return;
  const int rowTile = tile / colTiles;
  const int colTile = tile - rowTile * colTiles;
  const int row0 = rowTile << 4;
  const int col0 = colTile << 4;

  const float* aRow  = A + (size_t)(row0 + mr) * K + 2 * half;
  const float* bBase = ldsW + (size_t)(2 * half) * CO_PAD + col0 + mr;

  v8f acc = {};
#pragma unroll 4
  for (int k0 = 0; k0 < K; k0 += 4) {
    v2f a = *(const v2f*)(aRow + k0);      // global_load_b64 (8B aligned)
    const float* bp = bBase + (size_t)k0 * CO_PAD;
    v2f b;
    b.x = bp[0];
    b.y = bp[CO_PAD];
    acc = __builtin_amdgcn_wmma_f32_16x16x4_f32(
        /*neg_a=*/false, a, /*neg_b=*/false, b,
        /*c_mod=*/(short)0, acc, /*reuse_a=*/false, /*reuse_b=*/false);
  }

  if (CO == CO_PAD || (col0 + mr) < CO) {
    float* o = H + (size_t)(row0 + 8 * half) * CO + col0 + mr;
#pragma unroll
    for (int j = 0; j < 8; ++j) o[(size_t)j * CO] = acc[j];
  }
}

// ---------------------------- edge aggregation -----------------------------
// agg[dst] += h[src] * dinv[src]*dinv[dst], real edges only (self-loops fused
// into the bias pass). 16 threads per edge x float4 for the 64-wide layers.

__global__ __launch_bounds__(256) void k_aggregate64(const float* __restrict__ h,
                                                     const long long* __restrict__ src,
                                                     const long long* __restrict__ dst,
                                                     const float* __restrict__ dinv,
                                                     float* __restrict__ agg, int nE) {
  int i = blockIdx.x * blockDim.x + threadIdx.x;   // nE*16 = 12.8M
  int e = i >> 4;
  if (e >= nE) return;
  int c0 = (i & 15) << 2;
  int s = (int)src[e];
  int d = (int)dst[e];
  float w = dinv[s] * dinv[d];
  const float4 hv = *(const float4*)(h + (size_t)s * 64 + c0);
  float* o = agg + (size_t)d * 64 + c0;
  atomicAdd(o + 0, hv.x * w);
  atomicAdd(o + 1, hv.y * w);
  atomicAdd(o + 2, hv.z * w);
  atomicAdd(o + 3, hv.w * w);
}

__global__ __launch_bounds__(256) void k_aggregate40(const float* __restrict__ h,
                                                     const long long* __restrict__ src,
                                                     const long long* __restrict__ dst,
                                                     const float* __restrict__ dinv,
                                                     float* __restrict__ agg, int nE) {
  int i = blockIdx.x * blockDim.x + threadIdx.x;   // nE*40 = 32M
  if (i >= nE * 40) return;
  int e = i / 40;
  int c = i - e * 40;
  int s = (int)src[e];
  int d = (int)dst[e];
  float w = dinv[s] * dinv[d];
  atomicAdd(agg + (size_t)d * 40 + c, h[(size_t)s * 40 + c] * w);
}

// ------------------- self-loop + bias (+ReLU) fusion -----------------------

template <int C>
__global__ __launch_bounds__(256) void k_bias_self_relu(const float* __restrict__ agg,
                                                        const float* __restrict__ h,
                                                        const float* __restrict__ dinv,
                                                        const float* __restrict__ b,
                                                        float* __restrict__ out, int n) {
  int i = blockIdx.x * blockDim.x + threadIdx.x;   // n*C = 3.2M
  if (i >= n * C) return;
  int node = i / C;
  int c = i - node * C;
  float di = dinv[node];
  float v = agg[i] + h[i] * (di * di) + b[c];
  out[i] = fmaxf(v, 0.0f);
}

// --------------- final: self-loop + bias + log_softmax (wave32) ------------

__global__ __launch_bounds__(256) void k_final_logsoftmax(const float* __restrict__ agg,
                                                          const float* __restrict__ h,
                                                          const float* __restrict__ dinv,
                                                          const float* __restrict__ b,
                                                          float* __restrict__ out, int n) {
  int node = blockIdx.x * 8 + (threadIdx.x >> 5);
  int lane = threadIdx.x & 31;
  if (node >= n) return;
  float di = dinv[node];
  float dd = di * di;
  const float* ag = agg + (size_t)node * 40;
  const float* hh = h + (size_t)node * 40;
  // 40 columns on 32 lanes: lane covers c=lane, plus c=lane+32 for lanes 0..7
  float v0 = ag[lane] + hh[lane] * dd + b[lane];
  float v1 = -INFINITY;
  if (lane < 8) v1 = ag[lane + 32] + hh[lane + 32] * dd + b[lane + 32];
  float m = fmaxf(v0, v1);
#pragma unroll
  for (int off = 16; off > 0; off >>= 1) m = fmaxf(m, __shfl_xor(m, off));
  float s = __expf(v0 - m) + ((lane < 8) ? __expf(v1 - m) : 0.0f);
#pragma unroll
  for (int off = 16; off > 0; off >>= 1) s += __shfl_xor(s, off);
  float lse = m + __logf(s);
  float* o = out + (size_t)node * 40;
  o[lane] = v0 - lse;
  if (lane < 8) o[lane + 32] = v1 - lse;
}

// ---------------------------------------------------------------------------

extern "C" void kernel_launch(void* const* d_in, const int* in_sizes, int n_in,
                              void* d_out, int out_size, void* d_ws, size_t ws_size,
                              hipStream_t stream) {
  const float* x      = (const float*)d_in[0];
  const long long* ei = (const long long*)d_in[1];   // int64 edge_index [2, E]
  const float* W1 = (const float*)d_in[2];
  const float* b1 = (const float*)d_in[3];
  const float* W2 = (const float*)d_in[4];
  const float* b2 = (const float*)d_in[5];
  const float* W3 = (const float*)d_in[6];
  const float* b3 = (const float*)d_in[7];
  float* out = (float*)d_out;

  const int IN_C = 128, HID = 64, OUT_C = 40;
  const int N = in_sizes[0] / IN_C;   // 50000 (multiple of 16)
  const int E = in_sizes[1] / 2;      // 800000
  const long long* src = ei;
  const long long* dst = ei + E;

  // workspace layout
  char* ws = (char*)d_ws;
  auto carve = [&](size_t bytes) {
    void* p = (void*)ws;
    ws += (bytes + 255) & ~(size_t)255;
    return p;
  };
  float* dinv = (float*)carve((size_t)N * 4);
  float* h    = (float*)carve((size_t)N * HID * 4);
  float* agg  = (float*)carve((size_t)N * HID * 4);
  float* act  = (float*)carve((size_t)N * HID * 4);

  const int T = 256;
  auto blk = [](long long n, int t) { return (unsigned)((n + t - 1) / t); };

  // 1) symmetric gcn_norm: deg (with self-loop) -> dinv
  k_fill<<<blk(N, T), T, 0, stream>>>(dinv, 1.0f, N);
  k_deg_accum<<<blk(E, T), T, 0, stream>>>(dst, dinv, E);
  k_rsqrt_inplace<<<blk(N, T), T, 0, stream>>>(dinv, N);

  // ---- layer 1: x[N,128] @ W1[128,64] -> aggregate -> relu ----
  k_fill<<<blk((long long)N * HID, T), T, 0, stream>>>(agg, 0.0f, N * HID);
  {
    int tiles = (N >> 4) * (64 / 16);
    k_gemm_wmma<128, 64, 64><<<blk(tiles, 8), T, 0, stream>>>(x, W1, h, N);
  }
  k_aggregate64<<<blk((long long)E * 16, T), T, 0, stream>>>(h, src, dst, dinv, agg, E);
  k_bias_self_relu<64><<<blk((long long)N * 64, T), T, 0, stream>>>(agg, h, dinv, b1, act, N);

  // ---- layer 2: act[N,64] @ W2[64,64] ----
  k_fill<<<blk((long long)N * HID, T), T, 0, stream>>>(agg, 0.0f, N * HID);
  {
    int tiles = (N >> 4) * (64 / 16);
    k_gemm_wmma<64, 64, 64><<<blk(tiles, 8), T, 0, stream>>>(act, W2, h, N);
  }
  k_aggregate64<<<blk((long long)E * 16, T), T, 0, stream>>>(h, src, dst, dinv, agg, E);
  k_bias_self_relu<64><<<blk((long long)N * 64, T), T, 0, stream>>>(agg, h, dinv, b2, act, N);

  // ---- layer 3: act[N,64] @ W3[64,40] (padded to 48 cols in LDS) ----
  k_fill<<<blk((long long)N * OUT_C, T), T, 0, stream>>>(agg, 0.0f, N * OUT_C);
  {
    int tiles = (N >> 4) * (48 / 16);
    k_gemm_wmma<64, 40, 48><<<blk(tiles, 8), T, 0, stream>>>(act, W3, h, N);
  }
  k_aggregate40<<<blk((long long)E * 40, T), T, 0, stream>>>(h, src, dst, dinv, agg, E);

  // ---- final: self-loop + bias + log_softmax -> d_out ----
  k_final_logsoftmax<<<blk(N, 8), T, 0, stream>>>(agg, h, dinv, b3, out, N);

  (void)n_in; (void)out_size; (void)ws_size;
}